// HGINLayer_88648124991553
// MI455X (gfx1250) — compile-verified
//
#include <hip/hip_runtime.h>

typedef float v2f __attribute__((ext_vector_type(2)));
typedef float v8f __attribute__((ext_vector_type(8)));

#define D 128
#define LDA 132  // padded LDS row stride (floats) -> conflict-free A-fragment reads

__device__ __forceinline__ void atomic_fadd(float* p, float v) {
#if defined(__HIP_DEVICE_COMPILE__)
  unsafeAtomicAdd(p, v);   // native global_atomic_add_f32
#else
  atomicAdd(p, v);
#endif
}

// ---------------- elementwise: I = (1+eps[k]) * x ----------------
__global__ void init_scaled(const float* __restrict__ x, const float* __restrict__ eps,
                            int k, float* __restrict__ I, long n) {
  float s = 1.0f + eps[k];
  long stride = (long)gridDim.x * blockDim.x;
  for (long i = blockIdx.x * (long)blockDim.x + threadIdx.x; i < n; i += stride)
    I[i] = s * x[i];
}

// ---------------- edge gather + scatter-add (32 float4 chunks / edge) ----------------
__global__ void edge_aggregate(const float* __restrict__ xsrc,
                               const int* __restrict__ src, const int* __restrict__ dst,
                               float* __restrict__ I, int nEdges) {
  long total = (long)nEdges * 32;
  long stride = (long)gridDim.x * blockDim.x;
  for (long i = blockIdx.x * (long)blockDim.x + threadIdx.x; i < total; i += stride) {
    int e = (int)(i >> 5);
    int c = (int)(i & 31);
    int sI = src[e];
    int dI = dst[e];
    float4 v = ((const float4*)(xsrc + (long)sI * D))[c];
    float* dp = I + (long)dI * D + c * 4;
    atomic_fadd(dp + 0, v.x);
    atomic_fadd(dp + 1, v.y);
    atomic_fadd(dp + 2, v.z);
    atomic_fadd(dp + 3, v.w);
  }
}

// ---------------- f32 GEMM: Out[N x 128] = A[N x 128] @ W[128 x 128] ----------------
// Block: 256 threads (8 waves). Block tile: 64 rows x 128 cols.
// Wave w -> column tile w (16 cols); 4 row sub-tiles of 16; K stepped by 4
// via v_wmma_f32_16x16x4_f32.
__global__ __launch_bounds__(256)
void gemm128_wmma(const float* __restrict__ A, const float* __restrict__ W,
                  float* __restrict__ Out, int N) {
  __shared__ float sA[64 * LDA];  // 33 KB
  const int rowBase = blockIdx.x * 64;
  const int tid = threadIdx.x;

  // Cooperative load of 64x128 A tile (zero-padded rows past N).
#pragma unroll
  for (int i = 0; i < 8; ++i) {
    int chunk = tid + i * 256;     // 2048 float4 chunks total
    int r = chunk >> 5;            // 32 float4 per row
    int c4 = chunk & 31;
    int gr = rowBase + r;
    float4 v = make_float4(0.f, 0.f, 0.f, 0.f);
    if (gr < N) v = ((const float4*)(A + (long)gr * D))[c4];
    *((float4*)(sA + r * LDA + c4 * 4)) = v;
  }
  __syncthreads();

  const int wave = tid >> 5;       // 0..7 : column tile
  const int lane = tid & 31;
  const int half = lane >> 4;      // 0/1 : K-half per WMMA fragment layout
  const int l16  = lane & 15;
  const int col  = wave * 16 + l16;

  v8f acc0 = {}, acc1 = {}, acc2 = {}, acc3 = {};
#pragma unroll 4
  for (int k0 = 0; k0 < D; k0 += 4) {
    int kk = k0 + half * 2;
    // B fragment (4x16 tile of W): lanes 0-15 rows k0/k0+1, lanes 16-31 rows k0+2/k0+3
    v2f b;
    b.x = W[(long)kk * D + col];
    b.y = W[(long)(kk + 1) * D + col];
    // A fragments from LDS (b64 per lane, padded stride -> no bank conflicts)
    const float* pa = sA + kk;
    v2f a0 = *(const v2f*)(pa + ( 0 + l16) * LDA);
    v2f a1 = *(const v2f*)(pa + (16 + l16) * LDA);
    v2f a2 = *(const v2f*)(pa + (32 + l16) * LDA);
    v2f a3 = *(const v2f*)(pa + (48 + l16) * LDA);
    acc0 = __builtin_amdgcn_wmma_f32_16x16x4_f32(false, a0, false, b, (short)0, acc0, false, false);
    acc1 = __builtin_amdgcn_wmma_f32_16x16x4_f32(false, a1, false, b, (short)0, acc1, false, false);
    acc2 = __builtin_amdgcn_wmma_f32_16x16x4_f32(false, a2, false, b, (short)0, acc2, false, false);
    acc3 = __builtin_amdgcn_wmma_f32_16x16x4_f32(false, a3, false, b, (short)0, acc3, false, false);
  }

  // D layout: VGPR v holds rows (v, v+8) for lane halves; col = wave*16 + l16.
  v8f accs[4] = {acc0, acc1, acc2, acc3};
  if (rowBase + 64 <= N) {
    // Fast path: whole block tile in range -> straight-line stores, EXEC stays full.
    float* ob = Out + (long)(rowBase + half * 8) * D + col;
#pragma unroll
    for (int r = 0; r < 4; ++r) {
#pragma unroll
      for (int v = 0; v < 8; ++v)
        ob[(long)(r * 16 + v) * D] = accs[r][v];
    }
  } else {
#pragma unroll
    for (int r = 0; r < 4; ++r) {
#pragma unroll
      for (int v = 0; v < 8; ++v) {
        int gr = rowBase + r * 16 + v + half * 8;
        if (gr < N) Out[(long)gr * D + col] = accs[r][v];
      }
    }
  }
}

// ---------------- BatchNorm (training-mode batch stats) ----------------
__global__ void zero_f32(float* __restrict__ p, long n) {
  long stride = (long)gridDim.x * blockDim.x;
  for (long i = blockIdx.x * (long)blockDim.x + threadIdx.x; i < n; i += stride)
    p[i] = 0.f;
}

__global__ void bn_stats(const float* __restrict__ H, int N, float* __restrict__ stats) {
  int t = threadIdx.x;  // 128 threads = 128 columns
  float s = 0.f, q = 0.f;
  for (int i = blockIdx.x; i < N; i += gridDim.x) {
    float v = H[(long)i * D + t];
    s += v;
    q += v * v;
  }
  atomic_fadd(&stats[t], s);
  atomic_fadd(&stats[D + t], q);
}

__global__ void bn_finalize(const float* __restrict__ stats, int N,
                            const float* __restrict__ gamma, const float* __restrict__ beta,
                            float* __restrict__ ab) {
  int t = threadIdx.x;  // 128 threads
  float invN = 1.0f / (float)N;
  float mean = stats[t] * invN;
  float var = stats[D + t] * invN - mean * mean;  // biased variance (ddof=0)
  float a = gamma[t] * rsqrtf(var + 1e-5f);
  ab[t] = a;
  ab[D + t] = beta[t] - mean * a;
}

// Grid stride is a multiple of 128, so each thread's column is loop-invariant.
__global__ void bn_apply_relu(float* __restrict__ H, const float* __restrict__ ab, long n) {
  long start = blockIdx.x * (long)blockDim.x + threadIdx.x;
  long stride = (long)gridDim.x * blockDim.x;
  int c = (int)(start & (D - 1));
  float a = ab[c], b = ab[D + c];
  for (long i = start; i < n; i += stride) {
    float v = H[i] * a + b;
    H[i] = v > 0.f ? v : 0.f;
  }
}

__global__ void bn_apply_relu_accum(const float* __restrict__ H, const float* __restrict__ ab,
                                    float* __restrict__ out, long n) {
  long start = blockIdx.x * (long)blockDim.x + threadIdx.x;
  long stride = (long)gridDim.x * blockDim.x;
  int c = (int)(start & (D - 1));
  float a = ab[c], b = ab[D + c];
  for (long i = start; i < n; i += stride) {
    float v = H[i] * a + b;
    out[i] += v > 0.f ? v : 0.f;
  }
}

// ---------------- driver ----------------
extern "C" void kernel_launch(void* const* d_in, const int* in_sizes, int n_in,
                              void* d_out, int out_size, void* d_ws, size_t ws_size,
                              hipStream_t stream) {
  const float* x[3]   = {(const float*)d_in[0], (const float*)d_in[1], (const float*)d_in[2]};
  const int*   ei     = (const int*)d_in[3];    // [9,2,E]
  const float* W1     = (const float*)d_in[4];  // [3,128,128]
  // d_in[5] = b1: drops out exactly under training-mode BN (constant column shift)
  const float* g1     = (const float*)d_in[6];
  const float* be1    = (const float*)d_in[7];
  const float* W2     = (const float*)d_in[8];
  // d_in[9] = b2: same cancellation
  const float* g2     = (const float*)d_in[10];
  const float* be2    = (const float*)d_in[11];
  const float* eps    = (const float*)d_in[12];

  const int  NC[3]   = {200000, 3000, 20000};
  const long offs[3] = {0L, 200000L * D, 203000L * D};
  const int  E = 500000;

  float* out   = (float*)d_out;
  float* I     = (float*)d_ws;                  // 200000*128 floats (102.4 MB)
  float* H     = I + 200000L * D;               // 200000*128 floats (102.4 MB)
  float* stats = H + 200000L * D;               // 256 floats (sum | sumsq)
  float* ab    = stats + 256;                   // 256 floats (scale | shift)

  zero_f32<<<2048, 256, 0, stream>>>(out, (long)out_size);

  for (int k = 0; k < 9; ++k) {
    int s = k / 3;      // source type -> selects MLP weights
    int t = k % 3;      // destination type -> selects node set / output slot
    int Nt = NC[t];
    long nel = (long)Nt * D;
    const int* src = ei + (size_t)(2 * k + 0) * E;
    const int* dst = ei + (size_t)(2 * k + 1) * E;

    init_scaled<<<1024, 256, 0, stream>>>(x[t], eps, k, I, nel);
    edge_aggregate<<<4096, 256, 0, stream>>>(x[s], src, dst, I, E);

    // layer 1: GEMM -> BN stats -> BN+ReLU in place
    gemm128_wmma<<<(Nt + 63) / 64, 256, 0, stream>>>(I, W1 + (size_t)s * D * D, H, Nt);
    zero_f32<<<1, 256, 0, stream>>>(stats, 256);
    bn_stats<<<512, 128, 0, stream>>>(H, Nt, stats);
    bn_finalize<<<1, 128, 0, stream>>>(stats, Nt, g1 + s * D, be1 + s * D, ab);
    bn_apply_relu<<<1024, 256, 0, stream>>>(H, ab, nel);

    // layer 2: GEMM (reuse I) -> BN stats -> BN+ReLU accumulated into output
    gemm128_wmma<<<(Nt + 63) / 64, 256, 0, stream>>>(H, W2 + (size_t)s * D * D, I, Nt);
    zero_f32<<<1, 256, 0, stream>>>(stats, 256);
    bn_stats<<<512, 128, 0, stream>>>(I, Nt, stats);
    bn_finalize<<<1, 128, 0, stream>>>(stats, Nt, g2 + s * D, be2 + s * D, ab);
    bn_apply_relu_accum<<<1024, 256, 0, stream>>>(I, ab, out + offs[t], nel);
  }
}